// GCNBlock_44263932953221
// MI455X (gfx1250) — compile-verified
//
#include <hip/hip_runtime.h>
#include <hip/hip_bf16.h>

// ---------------------------------------------------------------------------
// GCN layer: out = relu( segment_sum( norm * (x@W)[src], dst ) + b )
// norm = deg^-1/2[src] * deg^-1/2[dst], with self-loops added.
// N=50000, E=800000, D=64.  Memory bound: scatter traffic lives in L2 (192MB).
// GEMM uses fp32 WMMA (V_WMMA_F32_16X16X4_F32) to keep full precision.
// ---------------------------------------------------------------------------

typedef __attribute__((ext_vector_type(2))) float v2f;
typedef __attribute__((ext_vector_type(8))) float v8f;

#define D_FEAT 64

// ---- degree: init with self-loop contribution -----------------------------
__global__ void init_deg_kernel(float* __restrict__ deg, int n) {
    int i = blockIdx.x * blockDim.x + threadIdx.x;
    if (i < n) deg[i] = 1.0f;                 // self-loop
}

// ---- degree: count incoming edges via f32 global atomics ------------------
__global__ void count_deg_kernel(const long long* __restrict__ dst,
                                 float* __restrict__ deg, int e) {
    int i = blockIdx.x * blockDim.x + threadIdx.x;
    if (i < e) atomicAdd(&deg[(int)dst[i]], 1.0f);
}

// ---- dis = deg^-1/2 (deg >= 1 always, so no zero guard needed) ------------
__global__ void rsqrt_kernel(const float* __restrict__ deg,
                             float* __restrict__ dis, int n) {
    int i = blockIdx.x * blockDim.x + threadIdx.x;
    if (i < n) dis[i] = rsqrtf(deg[i]);
}

// ---- xw = x @ W via fp32 WMMA ---------------------------------------------
// Block = 128 threads (4 waves). Block computes a 16x64 slab; wave w owns
// output columns [16w, 16w+16). K loop: 16 steps of V_WMMA_F32_16X16X4_F32.
// A tile staged in LDS with stride-68 padding (68 mod 64 banks = 4 ->
// lanes 0..15 land on distinct banks for the fragment reads).
#define LDS_STRIDE 68
__global__ void __launch_bounds__(128)
gemm_wmma_kernel(const float* __restrict__ x, const float* __restrict__ w,
                 float* __restrict__ xw) {
    __shared__ float ldsA[16 * LDS_STRIDE];

    const int tid  = threadIdx.x;
    const int wave = tid >> 5;
    const int lane = tid & 31;
    const int hl   = lane >> 4;      // which half of the wave
    const int lrow = lane & 15;
    const long baseElem = (long)blockIdx.x * 16 * D_FEAT;

    // Cooperative, coalesced load of the 16x64 fp32 A tile into padded LDS.
    {
        const int i0 = tid * 8;                  // 0..1023, 8 floats/thread
        const int r = i0 >> 6, c = i0 & 63;
        const float4* g4 = (const float4*)(x + baseElem + i0);
        float4 v0 = g4[0];
        float4 v1 = g4[1];
        float* p = &ldsA[r * LDS_STRIDE + c];
        *(float4*)(p)     = v0;
        *(float4*)(p + 4) = v1;
    }
    __syncthreads();

    const int nbase = wave * 16;                 // N-tile base for this wave
    v8f c = {};
    #pragma unroll
    for (int kk = 0; kk < 16; ++kk) {
        const int k0 = kk * 4 + 2 * hl;
        // A fragment (ISA 32-bit A 16x4 layout): row=lrow, K=k0 / k0+1
        v2f a, b;
        a.x = ldsA[lrow * LDS_STRIDE + k0];
        a.y = ldsA[lrow * LDS_STRIDE + k0 + 1];
        // B fragment (row-striped): col=nbase+lrow, K rows k0 / k0+1
        b.x = w[(k0)     * D_FEAT + nbase + lrow];
        b.y = w[(k0 + 1) * D_FEAT + nbase + lrow];
        c = __builtin_amdgcn_wmma_f32_16x16x4_f32(
                /*neg_a=*/false, a, /*neg_b=*/false, b,
                /*c_mod=*/(short)0, c, /*reuse_a=*/false, /*reuse_b=*/false);
    }

    // C/D layout: VGPR j holds M = j + 8*hl, N = lrow
    float* outp = xw + baseElem + nbase + lrow;
    #pragma unroll
    for (int j = 0; j < 8; ++j) {
        const int m = j + 8 * hl;
        outp[(long)m * D_FEAT] = c[j];
    }
}

// ---- scalar tail for rows not covered by 16-row tiles (none for N=50000) --
__global__ void gemm_tail_kernel(const float* __restrict__ x,
                                 const float* __restrict__ w,
                                 float* __restrict__ xw, int row0, int nrows) {
    int t = blockIdx.x * blockDim.x + threadIdx.x;
    if (t >= nrows * D_FEAT) return;
    int row = row0 + (t >> 6), col = t & 63;
    float acc = 0.0f;
    for (int k = 0; k < D_FEAT; ++k)
        acc += x[(long)row * D_FEAT + k] * w[k * D_FEAT + col];
    xw[(long)row * D_FEAT + col] = acc;
}

// ---- out = b + dis[i]^2 * xw[i]  (bias + self-loop message) ---------------
__global__ void init_out_kernel(const float* __restrict__ xw,
                                const float* __restrict__ dis,
                                const float* __restrict__ bias,
                                float* __restrict__ out, int n) {
    int t = blockIdx.x * blockDim.x + threadIdx.x;
    if (t >= n * D_FEAT) return;
    int i = t >> 6, j = t & 63;
    float d = dis[i];
    out[t] = bias[j] + d * d * xw[t];
}

// ---- edge scatter: 64 contiguous lanes per edge ---------------------------
// Gathers hit L2 (xw = 12.8MB << 192MB); atomics are contiguous f32 adds.
__global__ void scatter_kernel(const long long* __restrict__ src,
                               const long long* __restrict__ dst,
                               const float* __restrict__ xw,
                               const float* __restrict__ dis,
                               float* __restrict__ out, int e) {
    int t = blockIdx.x * blockDim.x + threadIdx.x;
    int eidx = t >> 6;
    if (eidx >= e) return;
    int j = t & 63;
    int s = (int)src[eidx];
    int d = (int)dst[eidx];
    float norm = dis[s] * dis[d];
    atomicAdd(&out[(long)d * D_FEAT + j], norm * xw[(long)s * D_FEAT + j]);
}

// ---- final ReLU -----------------------------------------------------------
__global__ void relu_kernel(float* __restrict__ out, int total) {
    int t = blockIdx.x * blockDim.x + threadIdx.x;
    if (t < total) out[t] = fmaxf(out[t], 0.0f);
}

// ---------------------------------------------------------------------------
extern "C" void kernel_launch(void* const* d_in, const int* in_sizes, int n_in,
                              void* d_out, int out_size, void* d_ws, size_t ws_size,
                              hipStream_t stream) {
    const float*     x  = (const float*)d_in[0];
    const long long* ei = (const long long*)d_in[1];   // int64 (2, E)
    const float*     W  = (const float*)d_in[2];
    const float*     b  = (const float*)d_in[3];
    float*           out = (float*)d_out;

    const int n = in_sizes[0] / D_FEAT;     // 50000
    const int e = in_sizes[1] / 2;          // 800000
    const long long* src = ei;
    const long long* dst = ei + e;

    // workspace layout: xw[n*64] | deg[n] | dis[n]
    char*  ws  = (char*)d_ws;
    float* xw  = (float*)ws;
    float* deg = (float*)(ws + (size_t)n * D_FEAT * sizeof(float));
    float* dis = deg + n;

    const int B = 256;
    init_deg_kernel<<<(n + B - 1) / B, B, 0, stream>>>(deg, n);
    count_deg_kernel<<<(e + B - 1) / B, B, 0, stream>>>(dst, deg, e);
    rsqrt_kernel<<<(n + B - 1) / B, B, 0, stream>>>(deg, dis, n);

    const int mtiles = n / 16;
    if (mtiles > 0)
        gemm_wmma_kernel<<<mtiles, 128, 0, stream>>>(x, W, xw);
    const int rem = n - mtiles * 16;
    if (rem > 0)
        gemm_tail_kernel<<<(rem * D_FEAT + B - 1) / B, B, 0, stream>>>(
            x, W, xw, mtiles * 16, rem);

    const int total = n * D_FEAT;           // 3.2M
    init_out_kernel<<<(total + B - 1) / B, B, 0, stream>>>(xw, dis, b, out, n);

    const long long escatter = (long long)e * D_FEAT;   // 51.2M threads
    scatter_kernel<<<(int)((escatter + B - 1) / B), B, 0, stream>>>(
        src, dst, xw, dis, out, e);

    relu_kernel<<<(total + B - 1) / B, B, 0, stream>>>(out, total);
}